// StructureSimilarityLoss_16346645528909
// MI455X (gfx1250) — compile-verified
//
#include <hip/hip_runtime.h>
#include <math.h>

typedef __attribute__((ext_vector_type(2))) float v2f;
typedef __attribute__((ext_vector_type(4))) float v4f;
typedef __attribute__((ext_vector_type(8))) float v8f;

#define H_IMG 512
#define W_IMG 512
#define PLANES 24                      // N*C = 8*3
#define REG 28                         // padded 26x26 halo region -> 28x28
#define N_TILES (PLANES * 32 * 32)     // 24576 16x16 tiles
#define WAVES_PER_BLOCK 4
#define NBLOCKS 768
#define TOTAL_WAVES (NBLOCKS * WAVES_PER_BLOCK)   // 3072
#define TILES_PER_WAVE (N_TILES / TOTAL_WAVES)    // 8 (exact)

// per-wave LDS slice layout (floats)
#define XS_OFF 0
#define YS_OFF (REG * REG)             // 784
#define HT_OFF (2 * REG * REG)         // 1568
#define HT_QS  (16 * REG)              // 448 floats per quantity (column-major 16 cols x 28 rows)
#define WAVE_LDS (HT_OFF + 5 * HT_QS)  // 3808 floats = 15232 B

// Gaussian(sigma=1.5), 11 taps, normalized
__device__ __forceinline__ float gw_at(int i) {
  float r = 0.0f;
  r = (i == 0)  ? 0.00102850f : r;
  r = (i == 1)  ? 0.00759875f : r;
  r = (i == 2)  ? 0.03600077f : r;
  r = (i == 3)  ? 0.10936070f : r;
  r = (i == 4)  ? 0.21300553f : r;
  r = (i == 5)  ? 0.26601172f : r;
  r = (i == 6)  ? 0.21300553f : r;
  r = (i == 7)  ? 0.10936070f : r;
  r = (i == 8)  ? 0.03600077f : r;
  r = (i == 9)  ? 0.00759875f : r;
  r = (i == 10) ? 0.00102850f : r;
  return r;
}

__device__ __forceinline__ v8f wmma4(v2f a, v2f b, v8f c) {
  // D = A(16x4,f32) * B(4x16,f32) + C(16x16,f32)
  return __builtin_amdgcn_wmma_f32_16x16x4_f32(false, a, false, b, (short)0, c,
                                               false, false);
}

__global__ __launch_bounds__(128, 1)
void ssim_struct_main(const float* __restrict__ x, const float* __restrict__ y,
                      float* __restrict__ partials) {
  __shared__ float lds[WAVES_PER_BLOCK * WAVE_LDS];
  __shared__ float bsum[WAVES_PER_BLOCK];

  const int tid  = threadIdx.x;
  const int wave = tid >> 5;
  const int lane = tid & 31;
  const int n16  = lane & 15;   // N index (and A-row index M)
  const int hi   = lane >> 4;   // half-wave select

  float* xs = lds + wave * WAVE_LDS + XS_OFF;
  float* ys = lds + wave * WAVE_LDS + YS_OFF;
  float* ht = lds + wave * WAVE_LDS + HT_OFF;

  // Gaussian fragments: serve as horizontal-B (G[K][N]=g[K-N]) and
  // vertical-A (G'[M][K]=g[K-M]) simultaneously (same per-lane value).
  v2f gfrag[7];
#pragma unroll
  for (int c = 0; c < 7; ++c) {
    const int j0 = 4 * c + 2 * hi;
    gfrag[c].x = gw_at(j0 - n16);
    gfrag[c].y = gw_at(j0 + 1 - n16);
  }

  float wsum = 0.0f;
  const int waveGlobal = blockIdx.x * WAVES_PER_BLOCK + wave;

#pragma unroll 1
  for (int t = 0; t < TILES_PER_WAVE; ++t) {
    const int tile  = waveGlobal + t * TOTAL_WAVES;
    const int plane = tile >> 10;      // / 1024 tiles per plane
    const int tin   = tile & 1023;
    const int tr    = tin >> 5;
    const int tc    = tin & 31;
    const float* xp = x + (size_t)plane * (H_IMG * W_IMG);
    const float* yp = y + (size_t)plane * (H_IMG * W_IMG);
    const int r0 = tr * 16 - 5;
    const int c0 = tc * 16 - 5;

    // ---- stage 28x28 zero-padded region of x and y into LDS ----
    if (lane < REG) {
      const int gc = c0 + lane;
      const bool cok = (lane < 26) & (gc >= 0) & (gc < W_IMG);
#pragma unroll 1
      for (int i = 0; i < REG; ++i) {
        const int gr = r0 + i;
        const bool ok = cok & (i < 26) & (gr >= 0) & (gr < H_IMG);
        float xv = 0.0f, yv = 0.0f;
        if (ok) {
          const size_t o = (size_t)gr * W_IMG + gc;
          xv = xp[o];
          yv = yp[o];
        }
        xs[i * REG + lane] = xv;
        ys[i * REG + lane] = yv;
      }
    }
    asm volatile("s_wait_dscnt 0" ::: "memory");

    // ---- horizontal pass: H(28x16) = Region(28x28) x G(28x16), 5 quantities ----
#pragma unroll
    for (int rbk = 0; rbk < 2; ++rbk) {
      const int rb = rbk * 12;  // row blocks 0..15 and 12..27 (overlap recomputed)
      v8f a0 = {}, a1 = {}, a2 = {}, a3 = {}, a4 = {};
#pragma unroll
      for (int c = 0; c < 7; ++c) {
        const int row = rb + n16;            // A-matrix M index
        const int col = 4 * c + 2 * hi;      // A-matrix K pair base
        const v2f ax = *(const v2f*)(xs + row * REG + col);  // ds_load_b64
        const v2f ay = *(const v2f*)(ys + row * REG + col);
        const v2f axy = ax * ay;
        const v2f axx = ax * ax;
        const v2f ayy = ay * ay;
        a0 = wmma4(ax,  gfrag[c], a0);
        a1 = wmma4(ay,  gfrag[c], a1);
        a2 = wmma4(axy, gfrag[c], a2);
        a3 = wmma4(axx, gfrag[c], a3);
        a4 = wmma4(ayy, gfrag[c], a4);
      }
      // store D column-major: lane owns column n16, rows rb+8*hi .. +7 (16B aligned)
      const int dbase = n16 * REG + rb + 8 * hi;
      {
        v4f lo, hq;
        float* d;
        d = ht + 0 * HT_QS + dbase;
        lo = (v4f){a0[0], a0[1], a0[2], a0[3]}; hq = (v4f){a0[4], a0[5], a0[6], a0[7]};
        *(v4f*)d = lo; *(v4f*)(d + 4) = hq;
        d = ht + 1 * HT_QS + dbase;
        lo = (v4f){a1[0], a1[1], a1[2], a1[3]}; hq = (v4f){a1[4], a1[5], a1[6], a1[7]};
        *(v4f*)d = lo; *(v4f*)(d + 4) = hq;
        d = ht + 2 * HT_QS + dbase;
        lo = (v4f){a2[0], a2[1], a2[2], a2[3]}; hq = (v4f){a2[4], a2[5], a2[6], a2[7]};
        *(v4f*)d = lo; *(v4f*)(d + 4) = hq;
        d = ht + 3 * HT_QS + dbase;
        lo = (v4f){a3[0], a3[1], a3[2], a3[3]}; hq = (v4f){a3[4], a3[5], a3[6], a3[7]};
        *(v4f*)d = lo; *(v4f*)(d + 4) = hq;
        d = ht + 4 * HT_QS + dbase;
        lo = (v4f){a4[0], a4[1], a4[2], a4[3]}; hq = (v4f){a4[4], a4[5], a4[6], a4[7]};
        *(v4f*)d = lo; *(v4f*)(d + 4) = hq;
      }
    }
    asm volatile("s_wait_dscnt 0" ::: "memory");

    // ---- vertical pass: Out(16x16) = G'(16x28) x H(28x16), 5 quantities ----
    v8f cx = {}, cy = {}, cxy = {}, cxx = {}, cyy = {};
#pragma unroll
    for (int c = 0; c < 7; ++c) {
      const int bbase = n16 * REG + 4 * c + 2 * hi;  // column-major H -> contiguous K pair
      const v2f b0 = *(const v2f*)(ht + 0 * HT_QS + bbase);
      const v2f b1 = *(const v2f*)(ht + 1 * HT_QS + bbase);
      const v2f b2 = *(const v2f*)(ht + 2 * HT_QS + bbase);
      const v2f b3 = *(const v2f*)(ht + 3 * HT_QS + bbase);
      const v2f b4 = *(const v2f*)(ht + 4 * HT_QS + bbase);
      cx  = wmma4(gfrag[c], b0, cx);
      cy  = wmma4(gfrag[c], b1, cy);
      cxy = wmma4(gfrag[c], b2, cxy);
      cxx = wmma4(gfrag[c], b3, cxx);
      cyy = wmma4(gfrag[c], b4, cyy);
    }

    // ---- per-pixel structure term + accumulate ----
#pragma unroll
    for (int v = 0; v < 8; ++v) {
      const float mux = cx[v], muy = cy[v];
      const float sxy = cxy[v] - mux * muy;
      const float sx  = fmaxf(cxx[v] - mux * mux, 1e-12f);
      const float sy  = fmaxf(cyy[v] - muy * muy, 1e-12f);
      const float st  = (sxy + 1e-4f) / (sqrtf(sx) * sqrtf(sy) + 1e-4f);
      wsum += st;
    }
    asm volatile("s_wait_dscnt 0" ::: "memory");  // before next tile reuses LDS
  }

  // wave32 reduction, then per-block partial
#pragma unroll
  for (int m = 16; m >= 1; m >>= 1) wsum += __shfl_xor(wsum, m, 32);
  if (lane == 0) bsum[wave] = wsum;
  __syncthreads();
  if (tid == 0)
    partials[blockIdx.x] = bsum[0] + bsum[1] + bsum[2] + bsum[3];
}

__global__ __launch_bounds__(256)
void ssim_struct_reduce(const float* __restrict__ partials, float* __restrict__ out) {
  __shared__ float s[256];
  float v = 0.0f;
  for (int i = threadIdx.x; i < NBLOCKS; i += 256) v += partials[i];
  s[threadIdx.x] = v;
  __syncthreads();
  for (int st = 128; st >= 1; st >>= 1) {
    if (threadIdx.x < st) s[threadIdx.x] += s[threadIdx.x + st];
    __syncthreads();
  }
  if (threadIdx.x == 0)
    out[0] = 1.0f - s[0] * (1.0f / (float)(PLANES * H_IMG * W_IMG));
}

extern "C" void kernel_launch(void* const* d_in, const int* in_sizes, int n_in,
                              void* d_out, int out_size, void* d_ws, size_t ws_size,
                              hipStream_t stream) {
  (void)in_sizes; (void)n_in; (void)out_size; (void)ws_size;
  const float* x = (const float*)d_in[0];
  const float* y = (const float*)d_in[1];
  float* out      = (float*)d_out;
  float* partials = (float*)d_ws;   // NBLOCKS floats, fully rewritten each call

  ssim_struct_main<<<NBLOCKS, 128, 0, stream>>>(x, y, partials);
  ssim_struct_reduce<<<1, 256, 0, stream>>>(partials, out);
}